// H2FDetector_layer_33191507263723
// MI455X (gfx1250) — compile-verified
//
#include <hip/hip_runtime.h>
#include <hip/hip_bf16.h>
#include <math.h>

// H2FDetector layer for MI455X (gfx1250, wave32).
// Pipeline:
//  K0: init out=0, denom=0, amax=-inf
//  K1: WMMA f32 16x16x4 node GEMMs: h = x@Wl+bl, hx = x@Wd+bd  (stored to ws)
//  K1b: per-node scalars d1,d2 (sign precursor) and a1,a2 (attention precursors)
//  K2: per-edge atomic-max of alpha into amax[dst,head] (int-punned float max)
//  K3: per-(edge,head): aexp = exp(alpha - amax); atomicAdd denom; atomicAdd out += aexp*sign*h[src]
//  K4: out /= max(denom, 1e-16)

typedef __attribute__((ext_vector_type(2))) float v2f;
typedef __attribute__((ext_vector_type(8))) float v8f;

#define NSLOPE 0.01f

// ---------------- K0: init ----------------
__global__ __launch_bounds__(256) void k0_init(float* __restrict__ out,
                                               float* __restrict__ denom,
                                               float* __restrict__ amax,
                                               int N) {
  int tid = blockIdx.x * blockDim.x + threadIdx.x;
  if (tid < N * 64) out[tid] = 0.0f;
  if (tid < N * 4) {
    denom[tid] = 0.0f;
    amax[tid] = -__builtin_inff();
  }
}

// ---------------- K1: WMMA node GEMMs ----------------
// One wave = one 16-node tile. Computes h (x@Wl+bl) and hx (x@Wd+bd), 16x64 each.
__global__ __launch_bounds__(256) void k1_node_gemm(
    const float* __restrict__ x, const float* __restrict__ Wl,
    const float* __restrict__ bl, const float* __restrict__ Wd,
    const float* __restrict__ bd, float* __restrict__ h,
    float* __restrict__ hx, int nTiles) {
  __shared__ float lw[2 * 64 * 64];  // Wl then Wd, 32 KB
  for (int i = threadIdx.x; i < 64 * 64; i += 256) {
    lw[i] = Wl[i];
    lw[4096 + i] = Wd[i];
  }
  __syncthreads();

  int wave = threadIdx.x >> 5;
  int lane = threadIdx.x & 31;
  int tile = blockIdx.x * 8 + wave;
  if (tile >= nTiles) return;  // no barriers after this point

  int half = lane >> 4;   // 0: lanes 0-15, 1: lanes 16-31
  int l16  = lane & 15;
  int arow = tile * 16 + l16;

  // A fragments for all 16 K-steps: lane needs x[arow][4k+2*half] and +1 (adjacent pair)
  float2 av[16];
  const float* xr = x + (size_t)arow * 64 + 2 * half;
#pragma unroll
  for (int k = 0; k < 16; ++k) av[k] = *(const float2*)(xr + 4 * k);

  const v8f vzero = {0.f, 0.f, 0.f, 0.f, 0.f, 0.f, 0.f, 0.f};
  v8f accL[4], accD[4];
#pragma unroll
  for (int j = 0; j < 4; ++j) { accL[j] = vzero; accD[j] = vzero; }

#pragma unroll
  for (int k = 0; k < 16; ++k) {
    v2f A;
    A[0] = av[k].x;  // K = 4k + 2*half     -> VGPR0 (K0 lower / K2 upper)
    A[1] = av[k].y;  // K = 4k + 1 + 2*half -> VGPR1 (K1 lower / K3 upper)
    int r0 = (4 * k + 2 * half) * 64;  // weight row for this lane-half
#pragma unroll
    for (int j = 0; j < 4; ++j) {
      int c = j * 16 + l16;  // lane = output column within tile
      v2f B;
      B[0] = lw[r0 + c];        // K = 4k+2*half
      B[1] = lw[r0 + 64 + c];   // K = 4k+1+2*half
      accL[j] = __builtin_amdgcn_wmma_f32_16x16x4_f32(
          false, A, false, B, (short)0, accL[j], false, false);
      v2f Bd;
      Bd[0] = lw[4096 + r0 + c];
      Bd[1] = lw[4096 + r0 + 64 + c];
      accD[j] = __builtin_amdgcn_wmma_f32_16x16x4_f32(
          false, A, false, Bd, (short)0, accD[j], false, false);
    }
  }

  // D layout: VGPR i -> row (i + 8*half), col = j*16 + l16
#pragma unroll
  for (int j = 0; j < 4; ++j) {
    int c = j * 16 + l16;
    float blv = bl[c], bdv = bd[c];
#pragma unroll
    for (int i = 0; i < 8; ++i) {
      size_t r = (size_t)tile * 16 + i + 8 * half;
      h[r * 64 + c]  = accL[j][i] + blv;
      hx[r * 64 + c] = accD[j][i] + bdv;
    }
  }
}

// ---------------- K1b: per-node scalars ----------------
__global__ __launch_bounds__(256) void k1b_node_scalars(
    const float* __restrict__ h, const float* __restrict__ hx,
    const float* __restrict__ Wa, const float* __restrict__ Wf,
    float* __restrict__ d1, float* __restrict__ d2,
    float* __restrict__ a1, float* __restrict__ a2, int N) {
  int n = blockIdx.x * blockDim.x + threadIdx.x;
  if (n >= N) return;
  const float* hr  = h + (size_t)n * 64;
  const float* hxr = hx + (size_t)n * 64;
  float s1 = 0.f, s2 = 0.f;
  float aa[4] = {0.f, 0.f, 0.f, 0.f};
  float ab[4] = {0.f, 0.f, 0.f, 0.f};
#pragma unroll 4
  for (int c = 0; c < 64; ++c) {
    float wfc = Wf[128 + c];
    float v = hxr[c];
    s1 = fmaf(v, Wf[c] + wfc, s1);        // hs coeff + (hs-hd) coeff
    s2 = fmaf(v, Wf[64 + c] - wfc, s2);   // hd coeff - (hs-hd) coeff
    float hv = hr[c];
    int head = c >> 4, t = c & 15;
    aa[head] = fmaf(hv, Wa[t], aa[head]);
    ab[head] = fmaf(hv, Wa[16 + t], ab[head]);
  }
  d1[n] = s1;
  d2[n] = s2;
#pragma unroll
  for (int j = 0; j < 4; ++j) {
    a1[n * 4 + j] = aa[j];
    a2[n * 4 + j] = ab[j];
  }
}

// Deterministic float atomic max via integer punning (native global_atomic_max_i32 / min_u32).
__device__ __forceinline__ void atomicMaxF(float* addr, float val) {
  if (val >= 0.0f)
    atomicMax((int*)addr, __float_as_int(val));
  else
    atomicMin((unsigned int*)addr, __float_as_uint(val));
}

__device__ __forceinline__ float signf(float v) {
  return (v > 0.f) ? 1.f : ((v < 0.f) ? -1.f : 0.f);
}
__device__ __forceinline__ float lrelu(float a) {
  return a > 0.f ? a : NSLOPE * a;
}

// ---------------- K2: per-edge segment max ----------------
__global__ __launch_bounds__(256) void k2_edge_max(
    const int* __restrict__ src, const int* __restrict__ dst,
    const float* __restrict__ d1, const float* __restrict__ d2,
    const float* __restrict__ a1, const float* __restrict__ a2,
    const float* __restrict__ bf, const float* __restrict__ ba,
    float* __restrict__ amax, int E) {
  int e = blockIdx.x * blockDim.x + threadIdx.x;
  if (e >= E) return;
  int s = src[e], d = dst[e];
  float sg = signf(d1[s] + d2[d] + bf[0]);
  float4 a1v = *(const float4*)(a1 + (size_t)s * 4);
  float4 a2v = *(const float4*)(a2 + (size_t)d * 4);
  float bav = ba[0];
  atomicMaxF(&amax[d * 4 + 0], lrelu(fmaf(sg, a1v.x, a2v.x + bav)));
  atomicMaxF(&amax[d * 4 + 1], lrelu(fmaf(sg, a1v.y, a2v.y + bav)));
  atomicMaxF(&amax[d * 4 + 2], lrelu(fmaf(sg, a1v.z, a2v.z + bav)));
  atomicMaxF(&amax[d * 4 + 3], lrelu(fmaf(sg, a1v.w, a2v.w + bav)));
}

// ---------------- K3: per-(edge,head) accumulate ----------------
__global__ __launch_bounds__(256) void k3_edge_accum(
    const int* __restrict__ src, const int* __restrict__ dst,
    const float* __restrict__ d1, const float* __restrict__ d2,
    const float* __restrict__ a1, const float* __restrict__ a2,
    const float* __restrict__ h, const float* __restrict__ bf,
    const float* __restrict__ ba, const float* __restrict__ amax,
    float* __restrict__ denom, float* __restrict__ out, int E) {
  int tid = blockIdx.x * blockDim.x + threadIdx.x;
  if (tid >= E * 4) return;
  int e = tid >> 2, head = tid & 3;
  int s = src[e], d = dst[e];
  float sg = signf(d1[s] + d2[d] + bf[0]);
  float al = lrelu(fmaf(sg, a1[(size_t)s * 4 + head], a2[(size_t)d * 4 + head] + ba[0]));
  float aexp = __expf(al - amax[d * 4 + head]);
  atomicAdd(&denom[d * 4 + head], aexp);
  float coef = aexp * sg;
  if (coef != 0.f) {
    const float4* hs = (const float4*)(h + (size_t)s * 64 + head * 16);
    float* op = out + (size_t)d * 64 + head * 16;
    float4 h0 = hs[0], h1 = hs[1], h2 = hs[2], h3 = hs[3];
    atomicAdd(&op[0],  coef * h0.x); atomicAdd(&op[1],  coef * h0.y);
    atomicAdd(&op[2],  coef * h0.z); atomicAdd(&op[3],  coef * h0.w);
    atomicAdd(&op[4],  coef * h1.x); atomicAdd(&op[5],  coef * h1.y);
    atomicAdd(&op[6],  coef * h1.z); atomicAdd(&op[7],  coef * h1.w);
    atomicAdd(&op[8],  coef * h2.x); atomicAdd(&op[9],  coef * h2.y);
    atomicAdd(&op[10], coef * h2.z); atomicAdd(&op[11], coef * h2.w);
    atomicAdd(&op[12], coef * h3.x); atomicAdd(&op[13], coef * h3.y);
    atomicAdd(&op[14], coef * h3.z); atomicAdd(&op[15], coef * h3.w);
  }
}

// ---------------- K4: normalize ----------------
__global__ __launch_bounds__(256) void k4_finalize(float* __restrict__ out,
                                                   const float* __restrict__ denom,
                                                   int N) {
  int tid = blockIdx.x * blockDim.x + threadIdx.x;
  if (tid >= N * 64) return;
  int n = tid >> 6;
  int head = (tid >> 4) & 3;
  out[tid] = out[tid] / fmaxf(denom[n * 4 + head], 1e-16f);
}

extern "C" void kernel_launch(void* const* d_in, const int* in_sizes, int n_in,
                              void* d_out, int out_size, void* d_ws, size_t ws_size,
                              hipStream_t stream) {
  const float* x  = (const float*)d_in[0];
  const int* src  = (const int*)d_in[1];
  const int* dst  = (const int*)d_in[2];
  const float* Wl = (const float*)d_in[3];
  const float* bl = (const float*)d_in[4];
  const float* Wa = (const float*)d_in[5];
  const float* ba = (const float*)d_in[6];
  const float* Wd = (const float*)d_in[7];
  const float* bd = (const float*)d_in[8];
  const float* Wf = (const float*)d_in[9];
  const float* bf = (const float*)d_in[10];
  float* out = (float*)d_out;

  const int N = in_sizes[0] / 64;
  const int E = in_sizes[1];

  // workspace layout (floats)
  float* ws    = (float*)d_ws;
  float* h     = ws;                        // N*64
  float* hx    = h  + (size_t)N * 64;       // N*64
  float* d1    = hx + (size_t)N * 64;       // N
  float* d2    = d1 + N;                    // N
  float* a1    = d2 + N;                    // N*4
  float* a2    = a1 + (size_t)N * 4;        // N*4
  float* amax  = a2 + (size_t)N * 4;        // N*4
  float* denom = amax + (size_t)N * 4;      // N*4

  const int B = 256;
  // K0: init out/denom/amax
  k0_init<<<(N * 64 + B - 1) / B, B, 0, stream>>>(out, denom, amax, N);
  // K1: WMMA node GEMMs (one wave per 16 nodes, 8 waves per block)
  int nTiles = (N + 15) / 16;
  k1_node_gemm<<<(nTiles + 7) / 8, B, 0, stream>>>(x, Wl, bl, Wd, bd, h, hx, nTiles);
  // K1b: per-node scalars
  k1b_node_scalars<<<(N + B - 1) / B, B, 0, stream>>>(h, hx, Wa, Wf, d1, d2, a1, a2, N);
  // K2: segment max of attention logits
  k2_edge_max<<<(E + B - 1) / B, B, 0, stream>>>(src, dst, d1, d2, a1, a2, bf, ba, amax, E);
  // K3: softmax-weighted scatter accumulate
  k3_edge_accum<<<(E * 4 + B - 1) / B, B, 0, stream>>>(src, dst, d1, d2, a1, a2, h, bf, ba,
                                                       amax, denom, out, E);
  // K4: normalize
  k4_finalize<<<(N * 64 + B - 1) / B, B, 0, stream>>>(out, denom, N);
}